// TensorProduct_21303037788562
// MI455X (gfx1250) — compile-verified
//
#include <hip/hip_runtime.h>

typedef __attribute__((ext_vector_type(2))) float v2f;
typedef __attribute__((ext_vector_type(4))) float v4f;
typedef __attribute__((ext_vector_type(8))) float v8f;

namespace {
constexpr int kB     = 8192;  // batch
constexpr int kL     = 9;     // L1 == L2
constexpr int kN     = 25;    // output l-dim
constexpr int kF     = 128;   // channels
constexpr int kK     = 81;    // contraction length (L1*L2)
constexpr int kKpad  = 84;    // 21 k-steps of 4
constexpr int kSteps = 21;
constexpr int kTStr  = 90;    // T row stride: even (b64 align), gcd(90,64)=2 -> conflict-free stores
constexpr int kCgStr = 92;    // cg row stride: 92%64=28 -> n-lanes hit distinct banks, halves disjoint
constexpr int kWaves = 2;     // 2 waves/block -> 57.9 KB static LDS
}

// GEMM D = T x CG via V_WMMA_F32_16X16X4_F32 (full f32 precision):
//   A (16x4):  lanes 0-15 -> M=lane,    K = 4j+{0,1}
//              lanes16-31 -> M=lane-16, K = 4j+{2,3}
//   B (4x16):  mirrored K mapping, N = lane&15
//   C/D:       VGPR v, half h -> M = v + 8h, N = lane&15
__global__ __launch_bounds__(64)
void tensor_product_wmma(const float* __restrict__ x1,
                         const float* __restrict__ x2,
                         const float* __restrict__ cg,
                         float* __restrict__ out)
{
  __shared__ float s_cg[32 * kCgStr];            // [n][k], zero padded
  __shared__ float s_T[kWaves][2 * 32 * kTStr];  // [channel][row][k]

  const int tid  = threadIdx.x;
  const int lane = tid & 31;
  const int wav  = tid >> 5;

  // ---- stage CG as [n][k] (cg memory layout is [(l*9+m)][n]) ----
  for (int i = tid; i < 32 * kCgStr; i += 64) s_cg[i] = 0.0f;
  __syncthreads();
  for (int i = tid; i < kK * kN; i += 64) {
    const int k = i / kN;
    const int n = i - k * kN;
    s_cg[n * kCgStr + k] = cg[i];
  }
  __syncthreads();

  // ---- each wave owns 32 rows = one batch b, 32 consecutive channels f ----
  const int w     = blockIdx.x * kWaves + wav;
  const int rg    = w * 32 + lane;   // global (b,f) row, lane <-> row bijection
  const int b     = rg >> 7;
  const int f     = rg & 127;
  const int fbase = (w * 32) & 127;

  const float* p1 = x1 + (size_t)b * (2 * kL * kF) + f;
  const float* p2 = x2 + (size_t)b * (2 * kL * kF) + f;

  float x1e[kL], x1o[kL], x2e[kL], x2o[kL], x1s[kL];
#pragma unroll
  for (int l = 0; l < kL; ++l) {
    x1e[l] = p1[l * kF];
    x1o[l] = p1[(kL + l) * kF];
    x2e[l] = p2[l * kF];
    x2o[l] = p2[(kL + l) * kF];
    x1s[l] = x1e[l] + x1o[l];
  }

  // ---- build pair tensors T0/T1 for this lane's row; b64 stores of (k,k+1) ----
  float* tb = &s_T[wav][0];
  const int rowOff = lane * kTStr;
#pragma unroll
  for (int kk = 0; kk < 80; kk += 2) {       // compile-time (l,m), pairs may straddle l-rows
    const int l1 = kk / kL,       m1 = kk % kL;
    const int l2 = (kk + 1) / kL, m2 = (kk + 1) % kL;
    v2f t0p, t1p;
    t0p.x = fmaf(x1e[l1], x2o[m1], x1o[l1] * x2e[m1]);  // eee+ooe channel
    t0p.y = fmaf(x1e[l2], x2o[m2], x1o[l2] * x2e[m2]);
    t1p.x = x1s[l1] * x2o[m1];                          // eoo+oeo channel
    t1p.y = x1s[l2] * x2o[m2];
    *(v2f*)&tb[rowOff + kk]                 = t0p;
    *(v2f*)&tb[32 * kTStr + rowOff + kk]    = t1p;
  }
  { // k = 80 (l=m=8) plus zero pad up to kKpad=84
    v2f t0p, t1p, zz;
    t0p.x = fmaf(x1e[8], x2o[8], x1o[8] * x2e[8]); t0p.y = 0.0f;
    t1p.x = x1s[8] * x2o[8];                       t1p.y = 0.0f;
    zz.x = 0.0f; zz.y = 0.0f;
    *(v2f*)&tb[rowOff + 80]              = t0p;
    *(v2f*)&tb[rowOff + 82]              = zz;
    *(v2f*)&tb[32 * kTStr + rowOff + 80] = t1p;
    *(v2f*)&tb[32 * kTStr + rowOff + 82] = zz;
  }
  // wave-private LDS region: same-wave DS ordering + compiler dscnt waits suffice.

  const int h   = lane >> 4;   // half selects K pair within a k-step
  const int rlo = lane & 15;

  v8f acc[8] = {};             // [f-tile t][channel c][n-tile nt]

  for (int j = 0; j < kSteps; ++j) {
    const int k0 = 4 * j + 2 * h;

    // B fragments: contiguous (k0,k0+1) pair, 8B aligned, conflict-free
    const v2f bf0 = *(const v2f*)&s_cg[rlo * kCgStr + k0];
    const v2f bf1 = *(const v2f*)&s_cg[(rlo + 16) * kCgStr + k0];

    // A fragments: single b64 each, lands directly in an aligned VGPR pair
    v2f af[4];
#pragma unroll
    for (int t = 0; t < 2; ++t)
#pragma unroll
      for (int c = 0; c < 2; ++c)
        af[t * 2 + c] =
            *(const v2f*)&tb[c * 32 * kTStr + (16 * t + rlo) * kTStr + k0];

#pragma unroll
    for (int t = 0; t < 2; ++t)
#pragma unroll
      for (int c = 0; c < 2; ++c) {
        const int i = t * 2 + c;
        acc[i * 2 + 0] = __builtin_amdgcn_wmma_f32_16x16x4_f32(
            false, af[i], false, bf0, (short)0, acc[i * 2 + 0], false, false);
        acc[i * 2 + 1] = __builtin_amdgcn_wmma_f32_16x16x4_f32(
            false, af[i], false, bf1, (short)0, acc[i * 2 + 1], false, false);
      }
  }

  // ---- store: lane owns column n = 16*nt + rlo, 8 consecutive f per v8f half ----
#pragma unroll
  for (int t = 0; t < 2; ++t)
#pragma unroll
    for (int c = 0; c < 2; ++c)
#pragma unroll
      for (int nt = 0; nt < 2; ++nt) {
        const int n = nt * 16 + rlo;
        if (n < kN) {
          const v8f d = acc[(t * 2 + c) * 2 + nt];
          const size_t o = ((((size_t)b * 2 + c) * kN + n) * kF)
                           + fbase + t * 16 + h * 8;
          const v4f lo = {d[0], d[1], d[2], d[3]};
          const v4f hi = {d[4], d[5], d[6], d[7]};
          *(v4f*)&out[o]     = lo;   // global_store_b128
          *(v4f*)&out[o + 4] = hi;   // global_store_b128
        }
      }
}

extern "C" void kernel_launch(void* const* d_in, const int* in_sizes, int n_in,
                              void* d_out, int out_size, void* d_ws, size_t ws_size,
                              hipStream_t stream) {
  const float* x1 = (const float*)d_in[0];
  const float* x2 = (const float*)d_in[1];
  const float* cg = (const float*)d_in[2];
  float* out = (float*)d_out;

  constexpr int totalRows    = kB * kF;     // 1,048,576 (b,f) rows
  constexpr int rowsPerBlock = kWaves * 32; // 64
  static_assert(totalRows % rowsPerBlock == 0, "exact tiling");

  tensor_product_wmma<<<dim3(totalRows / rowsPerBlock), dim3(64), 0, stream>>>(
      x1, x2, cg, out);
}